// MultiHeadModel_23098334118525
// MI455X (gfx1250) — compile-verified
//
#include <hip/hip_runtime.h>

// Problem constants (match reference)
#define B_N 16384
#define D_K 1024
#define T_N 20
#define C_N 100

typedef float v2f __attribute__((ext_vector_type(2)));
typedef float v4f __attribute__((ext_vector_type(4)));
typedef float v8f __attribute__((ext_vector_type(8)));
typedef int   v4i __attribute__((ext_vector_type(4)));

// Workspace layout (int units). Rows padded per-task to multiples of 64 so a
// 64-row group always belongs to a single task.
#define OFF_COUNTS   0
#define OFF_CURSOR   64
#define OFF_POFF     128
#define OFF_META     192
#define OFF_GRPTASK  256
#define MAXGRP_CAP   288                  // >= B/64 + T = 276
#define OFF_PPERM    (OFF_GRPTASK + MAXGRP_CAP)
#define PPERM_CAP    (B_N + T_N * 64)     // 17664 worst-case padded rows

#define KCHUNK 64
#define LSTR   68                         // LDS row stride (floats): 64+4 pad -> conflict-free b64 reads

// ---- CDNA5 async global->LDS path; fall back to ld+ds_store ----
#if defined(__has_builtin)
# if __has_builtin(__builtin_amdgcn_global_load_async_to_lds_b128)
#  define HAVE_ASYNC_B128 1
# endif
# if __has_builtin(__builtin_amdgcn_s_wait_asynccnt)
#  define HAVE_WAIT_ASYNC 1
# endif
#endif

#if defined(HAVE_ASYNC_B128)
typedef __attribute__((address_space(1))) v4i* gptr_v4i;
typedef __attribute__((address_space(3))) v4i* lptr_v4i;
#endif

__device__ __forceinline__ void cp_b128(const float* __restrict__ gsrc, float* ldst) {
#if defined(HAVE_ASYNC_B128)
    __builtin_amdgcn_global_load_async_to_lds_b128(
        (gptr_v4i)(v4i*)const_cast<float*>(gsrc),
        (lptr_v4i)(v4i*)ldst, 0, 0);
#else
    *(v4f*)ldst = *(const v4f*)gsrc;
#endif
}

__device__ __forceinline__ void wait_async() {
#if defined(HAVE_ASYNC_B128)
# if defined(HAVE_WAIT_ASYNC)
    __builtin_amdgcn_s_wait_asynccnt(0);
# else
    asm volatile("s_wait_asynccnt 0" ::: "memory");
# endif
#endif
}

// ---------------- Phase 1: bucket samples by task ----------------

__global__ void k_init(int* __restrict__ ws) {
    int i = blockIdx.x * blockDim.x + threadIdx.x;
    if (i < PPERM_CAP) ws[OFF_PPERM + i] = -1;
    if (i < T_N) { ws[OFF_COUNTS + i] = 0; ws[OFF_CURSOR + i] = 0; }
}

__global__ void k_count(const int* __restrict__ t, int* __restrict__ ws) {
    int i = blockIdx.x * blockDim.x + threadIdx.x;
    if (i < B_N) atomicAdd(&ws[OFF_COUNTS + t[i]], 1);
}

// Serial scan over T=20 tasks: offsets padded to 64-row groups + group->task map.
__global__ void k_scan(int* __restrict__ ws) {
    if (blockIdx.x == 0 && threadIdx.x == 0) {
        int run = 0;
        for (int tt = 0; tt < T_N; ++tt) {
            ws[OFF_POFF + tt] = run;
            int ng = (ws[OFF_COUNTS + tt] + 63) >> 6;
            int gbase = run >> 6;
            for (int j = 0; j < ng; ++j) ws[OFF_GRPTASK + gbase + j] = tt;
            run += ng << 6;
        }
        ws[OFF_META + 0] = run;          // padded total rows
    }
}

__global__ void k_fill(const int* __restrict__ t, int* __restrict__ ws) {
    int i = blockIdx.x * blockDim.x + threadIdx.x;
    if (i < B_N) {
        int tt  = t[i];
        int pos = atomicAdd(&ws[OFF_CURSOR + tt], 1);
        ws[OFF_PPERM + ws[OFF_POFF + tt] + pos] = i;
    }
}

// One K-chunk of WMMA work for NJ column tiles, EXEC all-1s throughout.
// xsBase = &xs[aRow*LSTR + kofs]; wshBase = &wsh[(j0*16+n)*LSTR + kofs].
template <int NJ>
__device__ __forceinline__ void chunk_mma(const float* xsBase, const float* wshBase, v8f* acc) {
#pragma unroll 4
    for (int k = 0; k < KCHUNK; k += 4) {
        v2f a = *(const v2f*)(xsBase + k);
#pragma unroll
        for (int jj = 0; jj < NJ; ++jj) {
            v2f bf = *(const v2f*)(wshBase + jj * (16 * LSTR) + k);
            acc[jj] = __builtin_amdgcn_wmma_f32_16x16x4_f32(
                /*neg_a=*/false, a, /*neg_b=*/false, bf,
                /*c_mod=*/(short)0, acc[jj], /*reuse_a=*/false, /*reuse_b=*/false);
        }
    }
}

// ---------------- Phase 2: LDS-staged per-task WMMA GEMM ----------------
// Block = 256 threads (8 wave32s) owns a 64-row x 112-col tile of one task.
// Per 64-float K-chunk: stage x-tile (64x64) and W-tile (112x64) into LDS
// (async-to-LDS when available), then wave (rowTile 0-3, colHalf 0-1) runs
// V_WMMA_F32_16X16X4_F32 out of LDS.

__global__ __launch_bounds__(256) void k_gemm(const float* __restrict__ x,
                                              const float* __restrict__ W,
                                              const float* __restrict__ bias,
                                              const int* __restrict__ ws,
                                              float* __restrict__ out) {
    __shared__ float xs[64 * LSTR];      // 17.4 KB
    __shared__ float wsh[112 * LSTR];    // 30.5 KB
    __shared__ int   rowsLds[64];

    const int group = blockIdx.x;
    const int paddedTotal = ws[OFF_META + 0];
    if (group * 64 >= paddedTotal) return;            // uniform per block
    const int task = ws[OFF_GRPTASK + group];
    const int tid  = threadIdx.x;

    if (tid < 64) rowsLds[tid] = ws[OFF_PPERM + group * 64 + tid];
    __syncthreads();

    // x staging: thread -> (row = tid&63, 16-float segment = tid>>6)
    const int xr = tid & 63;
    const int xq = tid >> 6;
    int rowG = rowsLds[xr]; if (rowG < 0) rowG = 0;   // pad rows -> row 0 (never stored)
    const float* __restrict__ xrow = x + (size_t)rowG * D_K + xq * 16;
    float* xdst = &xs[xr * LSTR + xq * 16];

    // W staging: threads 0..223 -> (col row = tid>>1, 32-float half = tid&1)
    const int wr = tid >> 1;
    const int wh = tid & 1;
    int wc = wr < C_N ? wr : (C_N - 1);               // clamp pad cols (never stored)
    const float* __restrict__ wrow = W + ((size_t)task * C_N + wc) * D_K + wh * 32;
    float* wdst = &wsh[wr * LSTR + wh * 32];
    const bool doW = (tid < 224);

    // compute coords: wave = (rowTile, colHalf); colHalf forced into an SGPR so
    // the 3-vs-4 tile choice is a scalar branch (WMMA requires EXEC all-1s).
    const int wave     = tid >> 5;
    const int lane     = tid & 31;
    const int rowTile  = wave & 3;
    const int colHalfU = __builtin_amdgcn_readfirstlane(wave >> 2); // 0: j=0..3, 1: j=4..6
    const int j0       = colHalfU ? 4 : 0;
    const int hi       = lane >> 4;
    const int n        = lane & 15;
    const int kofs     = 2 * hi;                      // ISA A/B f32 fragment K split
    const int aRow     = rowTile * 16 + n;            // A fragment: m = lane&15

    const float* xsBase  = &xs[aRow * LSTR + kofs];
    const float* wshBase = &wsh[(j0 * 16 + n) * LSTR + kofs];

    v8f acc[4];
#pragma unroll
    for (int j = 0; j < 4; ++j) { v8f z = {}; acc[j] = z; }

    for (int kb = 0; kb < D_K; kb += KCHUNK) {
        __syncthreads();                              // previous chunk consumed
#pragma unroll
        for (int i = 0; i < 4; ++i) cp_b128(xrow + kb + i * 4, xdst + i * 4);
        if (doW) {
#pragma unroll
            for (int i = 0; i < 8; ++i) cp_b128(wrow + kb + i * 4, wdst + i * 4);
        }
        wait_async();
        __syncthreads();                              // chunk visible to all waves

        if (colHalfU == 0) {                          // scalar branch, EXEC untouched
            chunk_mma<4>(xsBase, wshBase, acc);
        } else {
            chunk_mma<3>(xsBase, wshBase, acc);
        }
    }

    // Epilogue: D layout -> VGPR r holds m = 16*rowTile + r + 8*hi, col n = lane&15.
    const int nj = colHalfU ? 3 : 4;
#pragma unroll
    for (int jj = 0; jj < 4; ++jj) {
        if (jj < nj) {
            int c = (j0 + jj) * 16 + n;
            if (c < C_N) {
                float bv = bias[task * C_N + c];
#pragma unroll
                for (int r = 0; r < 8; ++r) {
                    int m   = rowTile * 16 + r + 8 * hi;
                    int row = rowsLds[m];
                    if (row >= 0) out[(size_t)row * C_N + c] = acc[jj][r] + bv;
                }
            }
        }
    }
}

// ---------------- Launch ----------------

extern "C" void kernel_launch(void* const* d_in, const int* in_sizes, int n_in,
                              void* d_out, int out_size, void* d_ws, size_t ws_size,
                              hipStream_t stream) {
    const float* x = (const float*)d_in[0];
    const int*   t = (const int*)d_in[1];
    const float* W = (const float*)d_in[2];
    const float* b = (const float*)d_in[3];
    float* out = (float*)d_out;
    int*   ws  = (int*)d_ws;

    k_init <<<(PPERM_CAP + 255) / 256, 256, 0, stream>>>(ws);
    k_count<<<(B_N + 255) / 256,       256, 0, stream>>>(t, ws);
    k_scan <<<1, 32, 0, stream>>>(ws);
    k_fill <<<(B_N + 255) / 256,       256, 0, stream>>>(t, ws);

    const int maxGroups = B_N / 64 + T_N;             // 276 worst-case groups
    k_gemm <<<maxGroups, 256, 0, stream>>>(x, W, b, ws, out);
}